// KSparse_45157286150621
// MI455X (gfx1250) — compile-verified
//
#include <hip/hip_runtime.h>
#include <stdint.h>

#define ROWLEN 8192
#define NT 256
#define VEC_ITERS (ROWLEN / (NT * 4)) // 8 b128 chunks per thread

typedef float vf4 __attribute__((ext_vector_type(4)));

// Monotonic key transform: unsigned compare order == float compare order.
__device__ __forceinline__ uint32_t f2key(uint32_t b) {
    return b ^ ((uint32_t)((int32_t)b >> 31) | 0x80000000u);
}
__device__ __forceinline__ float key2f(uint32_t u) {
    uint32_t s = (uint32_t)((int32_t)u >> 31);
    return __uint_as_float(u ^ ((~s) | 0x80000000u));
}

__global__ __launch_bounds__(NT) void topk_thresh_mask(const float* __restrict__ in,
                                                       float* __restrict__ out,
                                                       const int* __restrict__ kptr) {
    __shared__ __align__(16) uint32_t s_key[ROWLEN];   // 32 KB: one full row
    __shared__ uint32_t s_hist[256];
    __shared__ uint32_t s_prefix;
    __shared__ uint32_t s_rank;

    const uint32_t tid  = threadIdx.x;
    const uint64_t gsrc = (uint64_t)(uintptr_t)in + (uint64_t)blockIdx.x * (ROWLEN * 4u);
    const uint32_t ldsb = (uint32_t)(uintptr_t)(&s_key[0]); // low 32 bits = LDS byte addr

    // ---- Phase 0: CDNA5 async DMA global -> LDS (b128/lane, ASYNCcnt path) ----
#pragma unroll
    for (int it = 0; it < VEC_ITERS; ++it) {
        const uint32_t off = (uint32_t)(it * NT + tid) * 16u;
        const uint32_t la  = ldsb + off;
        const uint64_t ga  = gsrc + off;
        asm volatile("global_load_async_to_lds_b128 %0, %1, off"
                     :: "v"(la), "v"(ga) : "memory");
    }
    s_hist[tid] = 0u;                                   // overlaps with DMA
    if (tid == 0) { s_prefix = 0u; s_rank = (uint32_t)(*kptr); }
    asm volatile("s_wait_asynccnt 0" ::: "memory");
    __syncthreads();

    // ---- Phase 1: fused in-place key transform + level-0 (top byte) histogram ----
    {
        uint4* kv4 = (uint4*)s_key;
#pragma unroll
        for (int j = 0; j < VEC_ITERS; ++j) {
            uint4 v = kv4[j * NT + tid];
            v.x = f2key(v.x); v.y = f2key(v.y); v.z = f2key(v.z); v.w = f2key(v.w);
            kv4[j * NT + tid] = v;
            atomicAdd(&s_hist[v.x >> 24], 1u);
            atomicAdd(&s_hist[v.y >> 24], 1u);
            atomicAdd(&s_hist[v.z >> 24], 1u);
            atomicAdd(&s_hist[v.w >> 24], 1u);
        }
    }
    __syncthreads();
    if (tid == 0) {  // select level-0 digit: first bin (descending) with cum > rank
        const uint32_t rr = s_rank;
        uint32_t cum = 0, digit = 0;
        for (int b = 255; b >= 0; --b) {
            const uint32_t c = s_hist[b];
            if (cum + c > rr) { digit = (uint32_t)b; break; }
            cum += c;
        }
        s_rank   = rr - cum;
        s_prefix = digit;
    }
    __syncthreads();

    // ---- Phase 2: levels 1..3, plain ds_add histograms (few matching lanes) ----
    for (int level = 1; level < 4; ++level) {
        const int shift = 24 - 8 * level;          // 16, 8, 0
        const uint32_t psh = (uint32_t)(shift + 8); // 24, 16, 8
        s_hist[tid] = 0u;
        __syncthreads();
        const uint32_t pfx = s_prefix;
        const uint4* kv4 = (const uint4*)s_key;
#pragma unroll
        for (int j = 0; j < VEC_ITERS; ++j) {
            const uint4 v = kv4[j * NT + tid];
            const uint32_t keys[4] = {v.x, v.y, v.z, v.w};
#pragma unroll
            for (int e = 0; e < 4; ++e) {
                const uint32_t key = keys[e];
                if ((key >> psh) == pfx)
                    atomicAdd(&s_hist[(key >> shift) & 0xFFu], 1u);
            }
        }
        __syncthreads();
        if (tid == 0) {
            const uint32_t rr = s_rank;
            uint32_t cum = 0, digit = 0;
            for (int b = 255; b >= 0; --b) {
                const uint32_t c = s_hist[b];
                if (cum + c > rr) { digit = (uint32_t)b; break; }
                cum += c;
            }
            s_rank   = rr - cum;
            s_prefix = (pfx << 8) | digit;
        }
        __syncthreads();
    }

    // ---- Phase 3: strict-> mask vs reconstructed threshold, non-temporal b128 stores ----
    const float tval = key2f(s_prefix);
    vf4* orow = (vf4*)(out + (size_t)blockIdx.x * ROWLEN);
    const uint4* kv4 = (const uint4*)s_key;
#pragma unroll
    for (int j = 0; j < VEC_ITERS; ++j) {
        const uint4 v = kv4[j * NT + tid];
        vf4 o; float f;
        f = key2f(v.x); o.x = (f > tval) ? f : 0.0f;
        f = key2f(v.y); o.y = (f > tval) ? f : 0.0f;
        f = key2f(v.z); o.z = (f > tval) ? f : 0.0f;
        f = key2f(v.w); o.w = (f > tval) ? f : 0.0f;
        __builtin_nontemporal_store(o, &orow[j * NT + tid]);
    }
}

extern "C" void kernel_launch(void* const* d_in, const int* in_sizes, int n_in,
                              void* d_out, int out_size, void* d_ws, size_t ws_size,
                              hipStream_t stream) {
    const float* in  = (const float*)d_in[0];
    const int*   kp  = (const int*)d_in[1];
    float*       out = (float*)d_out;
    const int rows = out_size / ROWLEN; // 4096
    topk_thresh_mask<<<dim3(rows), dim3(NT), 0, stream>>>(in, out, kp);
}